// TextLoss_27487790694672
// MI455X (gfx1250) — compile-verified
//
#include <hip/hip_runtime.h>
#include <hip/hip_bf16.h>
#include <math.h>

// ---------------------------------------------------------------------------
// TextLoss for MI455X (gfx1250, wave32).
// Streaming-reduction problem: ~131 MB read once -> ~5.6 us at 23.3 TB/s.
// One fused b128-load pass + histogram-based OHEM selection + tiny finalize.
// Wave reductions are done on the matrix pipe with V_WMMA_F32_16X16X4_F32
// (block-diagonal ones-A reduces two quantities per WMMA).
// ---------------------------------------------------------------------------

#define BATCH 8
#define HDIM 640
#define WDIM 640
#define PIX (HDIM * WDIM)                 // 409600 pixels per sample
#define NBINS 32768                       // float bits >> 16 (values >= 0)
#define TPB 256
#define ITERS 4
#define PX_PER_BLOCK (TPB * 4 * ITERS)    // 4096
#define BLOCKS_PER_SAMPLE (PIX / PX_PER_BLOCK)  // 100

typedef float v2f __attribute__((ext_vector_type(2)));
typedef float v8f __attribute__((ext_vector_type(8)));

// Static device scratch (zeroed every launch -> deterministic).
__device__ unsigned g_hist_cnt[BATCH * NBINS];
__device__ float    g_hist_sum[BATCH * NBINS];
__device__ float    g_pos_sum[BATCH];
__device__ float    g_pos_cnt[BATCH];
__device__ float    g_per_sample[BATCH];
__device__ float    g_scal[4];            // [0]=cls_sum [1]=norm_sum [2]=angle_sum [3]=m_cnt

__global__ __launch_bounds__(TPB) void tl_zero() {
  int i = blockIdx.x * blockDim.x + threadIdx.x;
  if (i < BATCH * NBINS) { g_hist_cnt[i] = 0u; g_hist_sum[i] = 0.0f; }
  if (i < BATCH) { g_pos_sum[i] = 0.0f; g_pos_cnt[i] = 0.0f; g_per_sample[i] = 0.0f; }
  if (i < 4) g_scal[i] = 0.0f;
}

// Reduce two per-lane quantities across the wave with one WMMA:
//   A (16x4): A[m][k] = 1 iff (m<8) == (k<2)  -> lanes 0-7 and 24-31 hold 1.
//   B (4x16): vgpr0 = qe (rows 0,1), vgpr1 = qo (rows 2,3).
//   D rows 0-7  = qe[n] + qe[n+16];  rows 8-15 = qo[n] + qo[n+16].
// c[0]: lanes 0-15 hold qe half-sums, lanes 16-31 hold qo half-sums;
// a 4-step xor butterfly within each 16-lane half finishes the reduction.
// Returns: lanes 0-15 = total(qe), lanes 16-31 = total(qo).
__device__ __forceinline__ float wave_pair_sum(float qe, float qo, float amask) {
  v2f a; a.x = amask; a.y = amask;
  v2f bm; bm.x = qe; bm.y = qo;
  v8f c = {0.f, 0.f, 0.f, 0.f, 0.f, 0.f, 0.f, 0.f};
  c = __builtin_amdgcn_wmma_f32_16x16x4_f32(false, a, false, bm,
                                            (short)0, c, false, false);
  float s = c[0];
  s += __shfl_xor(s, 1, 32);
  s += __shfl_xor(s, 2, 32);
  s += __shfl_xor(s, 4, 32);
  s += __shfl_xor(s, 8, 32);
  return s;
}

__global__ __launch_bounds__(TPB) void tl_main(
    const float* __restrict__ fy, const float* __restrict__ tmask,
    const int* __restrict__ trm, const float* __restrict__ dist,
    const float* __restrict__ dirf, const float* __restrict__ wmat) {
  const int b = blockIdx.x / BLOCKS_PER_SAMPLE;
  const int chunk = blockIdx.x % BLOCKS_PER_SAMPLE;
  const int tid = threadIdx.x;

  const float* fp0 = fy + (size_t)(4 * b + 0) * PIX;
  const float* fp1 = fy + (size_t)(4 * b + 1) * PIX;
  const float* fp2 = fy + (size_t)(4 * b + 2) * PIX;
  const float* fp3 = fy + (size_t)(4 * b + 3) * PIX;
  const float* tmp_ = tmask + (size_t)b * PIX;
  const int*   trp  = trm   + (size_t)b * PIX;
  const float* dsp  = dist  + (size_t)b * PIX;
  const float* dxp  = dirf + (size_t)(2 * b + 0) * PIX;
  const float* dyp  = dirf + (size_t)(2 * b + 1) * PIX;
  const float* wpp  = wmat + (size_t)b * PIX;
  unsigned* hcnt = g_hist_cnt + (size_t)b * NBINS;
  float*    hsum = g_hist_sum + (size_t)b * NBINS;

  float q_cls = 0.f, q_psum = 0.f, q_pcnt = 0.f, q_norm = 0.f;
  float q_ang = 0.f, q_mcnt = 0.f, q_zero = 0.f;

  #pragma unroll
  for (int it = 0; it < ITERS; ++it) {
    const int px = chunk * PX_PER_BLOCK + it * (TPB * 4) + tid * 4;
    float4 a0 = *(const float4*)(fp0 + px);
    float4 a1 = *(const float4*)(fp1 + px);
    float4 a2 = *(const float4*)(fp2 + px);
    float4 a3 = *(const float4*)(fp3 + px);
    float4 tm4 = *(const float4*)(tmp_ + px);
    int4   tr4 = *(const int4*)(trp + px);
    float4 ds4 = *(const float4*)(dsp + px);
    float4 gx4 = *(const float4*)(dxp + px);
    float4 gy4 = *(const float4*)(dyp + px);
    float4 wm4 = *(const float4*)(wpp + px);
    float f0[4] = {a0.x, a0.y, a0.z, a0.w};
    float f1[4] = {a1.x, a1.y, a1.z, a1.w};
    float f2[4] = {a2.x, a2.y, a2.z, a2.w};
    float f3[4] = {a3.x, a3.y, a3.z, a3.w};
    float tm[4] = {tm4.x, tm4.y, tm4.z, tm4.w};
    int   tr[4] = {tr4.x, tr4.y, tr4.z, tr4.w};
    float ds[4] = {ds4.x, ds4.y, ds4.z, ds4.w};
    float gx[4] = {gx4.x, gx4.y, gx4.z, gx4.w};
    float gy[4] = {gy4.x, gy4.y, gy4.z, gy4.w};
    float wm[4] = {wm4.x, wm4.y, wm4.z, wm4.w};

    #pragma unroll
    for (int j = 0; j < 4; ++j) {
      // --- cls: masked BCE ---
      float p = fminf(fmaxf(f0[j], 1e-7f), 1.0f - 1e-7f);
      float bce = (tr[j] > 0) ? (-logf(p)) : (-log1pf(-p));
      q_cls += bce * tm[j];

      // --- dis: masked MSE -> histogram for OHEM ---
      float dd = f1[j] - ds[j];
      float pre = dd * dd * tm[j];
      if (ds[j] >= 1e-3f) {
        q_psum += pre; q_pcnt += 1.0f;
      } else if (pre == 0.0f) {
        q_zero += 1.0f;                       // bucket-0 fast path (no atomic)
      } else {
        unsigned bin = __float_as_uint(pre) >> 16;  // monotone for pre >= 0
        atomicAdd(&hcnt[bin], 1u);
        atomicAdd(&hsum[bin], pre);
      }

      // --- flux norm loss ---
      float gn = sqrtf(gx[j] * gx[j] + gy[j] * gy[j]) + 1e-6f;
      float gtx = gx[j] / gn, gty = gy[j] / gn;
      float ddx = f2[j] - gtx, ddy = f3[j] - gty;
      q_norm += wm[j] * 0.5f * (ddx * ddx + ddy * ddy) * tm[j];

      // --- flux angle loss ---
      float pn = sqrtf(f2[j] * f2[j] + f3[j] * f3[j]) + 1e-6f;
      float pxn = f2[j] / pn, pyn = f3[j] / pn;
      float na = sqrtf(pxn * pxn + pyn * pyn);
      float nb = sqrtf(gtx * gtx + gty * gty);
      float cosv = (pxn * gtx + pyn * gty) / fmaxf(na * nb, 1e-8f);
      if (tm[j] > 0.0f && tr[j] > 0) { q_ang += 1.0f - cosv; q_mcnt += 1.0f; }
    }
  }

  // Wave reduction on the matrix pipe (EXEC all-ones here: full blocks, no
  // divergence at this point). Two quantities per WMMA.
  const int lane = tid & 31;
  const float amask = (((lane >> 3) ^ (lane >> 4)) & 1) ? 0.0f : 1.0f;
  float r0 = wave_pair_sum(q_cls,  q_psum, amask);
  float r1 = wave_pair_sum(q_pcnt, q_norm, amask);
  float r2 = wave_pair_sum(q_ang,  q_mcnt, amask);
  float r3 = wave_pair_sum(q_zero, 0.0f,   amask);
  if (lane == 0)  atomicAdd(&g_scal[0], r0);        // cls sum
  if (lane == 16) atomicAdd(&g_pos_sum[b], r0);     // pos sum
  if (lane == 0)  atomicAdd(&g_pos_cnt[b], r1);     // pos count
  if (lane == 16) atomicAdd(&g_scal[1], r1);        // norm sum
  if (lane == 0)  atomicAdd(&g_scal[2], r2);        // angle sum
  if (lane == 16) atomicAdd(&g_scal[3], r2);        // mask count
  if (lane == 0 && r3 > 0.5f)
    atomicAdd(&hcnt[0], (unsigned)(r3 + 0.5f));     // zeros -> bucket 0
}

// Per-sample OHEM finalize: cooperative suffix scan of the 32768-bin
// histogram to find the top-k threshold bin; exact sum above it, tie bin
// resolved by its average (exact in the common case k == n_neg).
__global__ __launch_bounds__(256) void tl_finalize() {
  const int b = blockIdx.x;
  const int t = threadIdx.x;           // 256 threads, 128 bins each
  __shared__ unsigned s_lcnt[256];
  __shared__ float s_lsum[256];
  const unsigned* cnt = g_hist_cnt + (size_t)b * NBINS;
  const float* hsum = g_hist_sum + (size_t)b * NBINS;

  const int base = t * 128;
  unsigned lc = 0; float ls = 0.0f;
  for (int i = 0; i < 128; ++i) { lc += cnt[base + i]; ls += hsum[base + i]; }
  s_lcnt[t] = lc; s_lsum[t] = ls;
  __syncthreads();

  unsigned cnt_above = 0; float sum_above = 0.0f;
  for (int u = t + 1; u < 256; ++u) { cnt_above += s_lcnt[u]; sum_above += s_lsum[u]; }

  unsigned n_pos = (unsigned)(g_pos_cnt[b] + 0.5f);
  unsigned n_neg = (unsigned)PIX - n_pos;
  unsigned k3 = 3u * n_pos;
  unsigned k = (n_pos > 0) ? ((n_neg < k3) ? n_neg : k3)
                           : ((n_neg < 100u) ? n_neg : 100u);
  float posi = g_pos_sum[b] / (float)((n_pos > 0u) ? n_pos : 1u);

  if (k == 0u) {
    if (t == 0) g_per_sample[b] = (n_pos > 0u) ? posi : 0.0f;
    return;
  }
  // Exactly one thread's range contains the crossing of the suffix count at k.
  if (cnt_above < k && cnt_above + lc >= k) {
    unsigned running = cnt_above; float rsum = sum_above;
    float nega_sum = 0.0f;
    for (int i = 127; i >= 0; --i) {
      unsigned c = cnt[base + i];
      if (running + c >= k) {
        unsigned need = k - running;                  // 0 < need <= c
        float avg = hsum[base + i] / (float)c;
        nega_sum = rsum + avg * (float)need;
        break;
      }
      running += c; rsum += hsum[base + i];
    }
    float nega = nega_sum / (float)k;
    g_per_sample[b] = (n_pos > 0u) ? (posi + nega) : nega;
  }
}

__global__ void tl_combine(float* __restrict__ out) {
  float cls = g_scal[0] / (float)((size_t)BATCH * PIX);
  float nrm = g_scal[1] / (float)(BATCH * HDIM);     // .sum(-1).mean()
  float ang = g_scal[2] / fmaxf(g_scal[3], 1.0f);
  float dis = 0.0f;
  #pragma unroll
  for (int b = 0; b < BATCH; ++b) dis += g_per_sample[b];
  dis /= (float)BATCH;
  out[0] = cls + 3.0f * dis + 0.5f * (nrm + ang);
}

extern "C" void kernel_launch(void* const* d_in, const int* in_sizes, int n_in,
                              void* d_out, int out_size, void* d_ws, size_t ws_size,
                              hipStream_t stream) {
  (void)in_sizes; (void)n_in; (void)out_size; (void)d_ws; (void)ws_size;
  const float* fy    = (const float*)d_in[0];   // [8,4,640,640]
  const float* tmask = (const float*)d_in[1];   // [8,640,640]
  const int*   trm   = (const int*)d_in[2];     // [8,640,640]
  const float* dist  = (const float*)d_in[3];   // [8,640,640]
  const float* dirf  = (const float*)d_in[4];   // [8,2,640,640]
  const float* wmat  = (const float*)d_in[5];   // [8,640,640]

  tl_zero<<<(BATCH * NBINS + TPB - 1) / TPB, TPB, 0, stream>>>();
  tl_main<<<BATCH * BLOCKS_PER_SAMPLE, TPB, 0, stream>>>(fy, tmask, trm, dist, dirf, wmat);
  tl_finalize<<<BATCH, 256, 0, stream>>>();
  tl_combine<<<1, 1, 0, stream>>>((float*)d_out);
}